// MessagePassing_15015205667414
// MI455X (gfx1250) — compile-verified
//
#include <hip/hip_runtime.h>
#include <math.h>

// Problem constants (reference: N=1, H=320, W=640, C=32, 4 dirs, 3 iters).
namespace {
constexpr int kH = 320, kW = 640, kC = 32;
constexpr int kHW = kH * kW;
constexpr float kFILL = 1e9f;
constexpr int kWavesPerBlock = 8;           // 8 pixels per 256-thread block
constexpr int kBlock = kWavesPerBlock * 32; // wave32
}

// ---- wave32 butterfly reductions (lane == label) -------------------------
__device__ __forceinline__ float wave_min32(float v) {
#pragma unroll
  for (int m = 16; m >= 1; m >>= 1) v = fminf(v, __shfl_xor(v, m, 32));
  return v;
}
__device__ __forceinline__ float wave_max32(float v) {
#pragma unroll
  for (int m = 16; m >= 1; m >>= 1) v = fmaxf(v, __shfl_xor(v, m, 32));
  return v;
}
__device__ __forceinline__ float wave_sum32(float v) {
#pragma unroll
  for (int m = 16; m >= 1; m >>= 1) v += __shfl_xor(v, m, 32);
  return v;
}

// One BP iteration (receiver-centric). LAST also emits softmax beliefs.
template <bool LAST>
__global__ __launch_bounds__(kBlock) void bp_iter(
    const float* __restrict__ prob,    // (H,W,C)
    const float* __restrict__ edge,    // (4,H,W)
    const float* __restrict__ aff,     // (4,3,H,W)
    const float* __restrict__ offs,    // (4,H,W)
    const float* __restrict__ msg_in,  // (4,H,W,C)  t
    float* __restrict__ msg_out,       // (4,H,W,C)  t+1
    float* __restrict__ beliefs) {     // (H,W,C) if LAST
  __shared__ float spar[kWavesPerBlock * 20];

  const int tid = threadIdx.x;
  const int wv = tid >> 5;
  const int lane = tid & 31;            // label index
  const int p = blockIdx.x * kWavesPerBlock + wv;  // pixel (grid divides evenly)
  const int y = p / kW;
  const int x = p - y * kW;

  // ---- Stage the 20 per-pixel scalars {edge, aff0, aff1, aff2, offset}×4dirs
  // into LDS. On gfx1250: one CDNA5 async global->LDS scatter per wave
  // (global_load_async_to_lds_b32, tracked by ASYNCcnt), else plain loads.
  {
    const int pd = (lane / 5) & 3;
    const int pk = lane % 5;
    const float* g;
    if (pk == 0)      g = edge + pd * kHW + p;
    else if (pk == 4) g = offs + pd * kHW + p;
    else              g = aff + (pd * 3 + (pk - 1)) * kHW + p;
#if __has_builtin(__builtin_amdgcn_global_load_async_to_lds_b32)
    typedef __attribute__((address_space(1))) int GInt;  // "__device__ int *"
    typedef __attribute__((address_space(3))) int LInt;  // "__shared__ int *"
    if (lane < 20) {
      __builtin_amdgcn_global_load_async_to_lds_b32(
          (GInt*)g, (LInt*)&spar[wv * 20 + lane], 0, 0);
    }
#if __has_builtin(__builtin_amdgcn_s_wait_asynccnt)
    __builtin_amdgcn_s_wait_asynccnt(0);
#else
    asm volatile("s_wait_asynccnt 0" ::: "memory");
#endif
    asm volatile("" ::: "memory");  // keep LDS reads below the wait
#else
    if (lane < 20) spar[wv * 20 + lane] = *g;
#endif
  }
  // Only this wave reads its own 20 params -> no block barrier needed.
  float par[20];
#pragma unroll
  for (int j = 0; j < 20; ++j) par[j] = spar[wv * 20 + j];

  float msum_new = 0.f;

#pragma unroll
  for (int d = 0; d < 4; ++d) {
    // d=0: sender x-1 (opp=1); d=1: sender x+1 (opp=0);
    // d=2: sender y-1 (opp=3); d=3: sender y+1 (opp=2).
    const bool valid = (d == 0) ? (x > 0)
                     : (d == 1) ? (x < kW - 1)
                     : (d == 2) ? (y > 0)
                                : (y < kH - 1);   // wave-uniform predicate
    float m = 0.f;                                 // boundary -> zero message
    if (valid) {
      const int sxx = (d == 0) ? x - 1 : (d == 1) ? x + 1 : x;
      const int syy = (d == 2) ? y - 1 : (d == 3) ? y + 1 : y;
      const int s = syy * kW + sxx;
      const float* mb = msg_in + (size_t)s * kC + lane;
      const float m0 = mb[0];
      const float m1 = mb[(size_t)1 * kHW * kC];
      const float m2 = mb[(size_t)2 * kHW * kC];
      const float m3 = mb[(size_t)3 * kHW * kC];
      const float cst = -prob[(size_t)s * kC + lane];  // cost = -prob_vol
      const float mo = (d == 0) ? m1 : (d == 1) ? m0 : (d == 2) ? m3 : m2;
      const float hs = cst + ((m0 + m1) + (m2 + m3)) - mo;

      // Label-shifted variants via wave32 lane shuffles (C==32==wave).
      float hm1 = __shfl_up(hs, 1, 32);   // hs[l-1]
      if (lane == 0) hm1 = kFILL;
      float hp1 = __shfl_down(hs, 1, 32); // hs[l+1]
      if (lane == 31) hp1 = kFILL;
      const float hmin = wave_min32(hs);

      const float w   = par[d * 5 + 0];
      const float a0  = par[d * 5 + 1];
      const float a1  = par[d * 5 + 2];
      const float l2  = par[d * 5 + 3];
      const float off = par[d * 5 + 4];
      const float a_m1 = (d & 1) ? a1 : a0;  // odd dirs swap +-1 jump costs
      const float a_p1 = (d & 1) ? a0 : a1;

      m = fminf(fminf(hs, hm1 + w * a_p1),
                fminf(hp1 + w * a_m1, hmin + w * l2)) + w * off;
      m -= wave_min32(m);  // message rescaling
    }
    msg_out[((size_t)d * kHW + p) * kC + lane] = m;
    if (LAST) msum_new += m;
  }

  if (LAST) {
    // beliefs = softmax(-(cost + sum(msg))) = softmax(prob - sum(msg))
    const float b = prob[(size_t)p * kC + lane] - msum_new;
    const float bm = wave_max32(b);
    const float e = expf(b - bm);
    const float es = wave_sum32(e);
    beliefs[(size_t)p * kC + lane] = e / es;
  }
}

extern "C" void kernel_launch(void* const* d_in, const int* in_sizes, int n_in,
                              void* d_out, int out_size, void* d_ws, size_t ws_size,
                              hipStream_t stream) {
  (void)in_sizes; (void)n_in; (void)out_size; (void)ws_size;
  const float* prob = (const float*)d_in[0];
  const float* edge = (const float*)d_in[1];
  const float* aff  = (const float*)d_in[2];
  const float* offs = (const float*)d_in[3];
  const float* msg0 = (const float*)d_in[4];

  float* beliefs  = (float*)d_out;                       // (H,W,C)
  float* out_msgs = beliefs + (size_t)kHW * kC;          // (4,H,W,C)
  float* ws_msgs  = (float*)d_ws;                        // (4,H,W,C) scratch

  dim3 grid(kHW / kWavesPerBlock), block(kBlock);
  // Jacobi double-buffering with only one scratch message volume:
  // iter1: d_in -> d_out(msgs); iter2: d_out(msgs) -> ws; iter3: ws -> d_out(msgs)+beliefs
  bp_iter<false><<<grid, block, 0, stream>>>(prob, edge, aff, offs, msg0,     out_msgs, nullptr);
  bp_iter<false><<<grid, block, 0, stream>>>(prob, edge, aff, offs, out_msgs, ws_msgs,  nullptr);
  bp_iter<true ><<<grid, block, 0, stream>>>(prob, edge, aff, offs, ws_msgs,  out_msgs, beliefs);
}